// GraphBackboneUpdate_64536178589748
// MI455X (gfx1250) — compile-verified
//
#include <hip/hip_runtime.h>

// GraphBackboneUpdate — memory-bound streaming kernel (AI ~0.5 flop/byte).
// Edge scoring dot-products done with exact-precision fp32 WMMA
// (V_WMMA_F32_16X16X4_F32), one wave per node, edge_h streamed once (NT).

typedef float v2f __attribute__((ext_vector_type(2)));
typedef float v8f __attribute__((ext_vector_type(8)));

#define KNB 32
#define DEDGE 128
#define DNODE 128
#define WAVES_PER_BLOCK 8

__global__ __launch_bounds__(256) void graph_backbone_kernel(
    const float* __restrict__ X,
    const float* __restrict__ node_h,
    const float* __restrict__ edge_h,
    const long long* __restrict__ edge_idx,
    const float* __restrict__ mask_ij,
    const float* __restrict__ Wu, const float* __restrict__ bu,
    const float* __restrict__ Wr, const float* __restrict__ br,
    float* __restrict__ out, int N)
{
    __shared__ __align__(16) float lds_wu[DEDGE];
    __shared__ __align__(16) float lds_wr[DNODE];
    __shared__ float lds_dx[WAVES_PER_BLOCK * 32];

    const int tid = threadIdx.x;
    if (tid < DEDGE) { lds_wu[tid] = Wu[tid]; lds_wr[tid] = Wr[tid]; }
    __syncthreads();

    const int lane   = tid & 31;
    const int waveid = tid >> 5;
    const int gwave  = blockIdx.x * WAVES_PER_BLOCK + waveid;
    const int nwaves = gridDim.x * WAVES_PER_BLOCK;
    const float bu0 = bu[0];
    const float br0 = br[0];

    // A-matrix 16x4 f32 layout (ISA 7.12.2): lanes 0-15 hold M=0..15 K={0,1},
    // lanes 16-31 hold M=0..15 K={2,3}.  B 4x16 is column-constant (Wu chunk),
    // so each lane's B fragment is just the matching K-pair of Wu.
    const int hi   = lane >> 4;   // selects K-pair {0,1} vs {2,3}
    const int mrow = lane & 15;   // M within tile
    float* dxbuf = lds_dx + waveid * 32;
    const float* wub = lds_wu + hi * 2;

    for (int n = gwave; n < N; n += nwaves) {
        const float* eb = edge_h + (size_t)n * (KNB * DEDGE);

        // warm next node's edge tile (global_prefetch_b8), 128B per lane = 4KB
        if (n + nwaves < N) {
            const float* pf = edge_h + (size_t)(n + nwaves) * (KNB * DEDGE);
            __builtin_prefetch(pf + lane * 32, 0, 1);
        }

        const float* row0 = eb + mrow * DEDGE + hi * 2;   // tile 0: k = 0..15
        const float* row1 = row0 + 16 * DEDGE;            // tile 1: k = 16..31

        v8f acc0 = {0.f, 0.f, 0.f, 0.f, 0.f, 0.f, 0.f, 0.f};
        v8f acc1 = {0.f, 0.f, 0.f, 0.f, 0.f, 0.f, 0.f, 0.f};
        #pragma unroll
        for (int c = 0; c < 32; ++c) {   // 32 chunks of K=4 cover DEDGE=128
            v2f a0 = __builtin_nontemporal_load((const v2f*)(row0 + c * 4));
            v2f a1 = __builtin_nontemporal_load((const v2f*)(row1 + c * 4));
            v2f bf = *(const v2f*)(wub + c * 4);          // ds_load broadcast
            acc0 = __builtin_amdgcn_wmma_f32_16x16x4_f32(
                       false, a0, false, bf, (short)0, acc0, false, false);
            acc1 = __builtin_amdgcn_wmma_f32_16x16x4_f32(
                       false, a1, false, bf, (short)0, acc1, false, false);
        }

        // D columns are identical; column-0 lanes (0 and 16) scatter the 32
        // dot products to LDS so lane k can pick up dx[k].
        if (mrow == 0) {
            const int o = hi * 8;        // lane 0 -> rows 0..7, lane 16 -> 8..15
            #pragma unroll
            for (int r = 0; r < 8; ++r) {
                dxbuf[o + r]      = acc0[r];   // k = o + r
                dxbuf[o + 16 + r] = acc1[r];   // k = 16 + o + r
            }
        }
        __builtin_amdgcn_wave_barrier();   // LDS pipe is in-order per wave
        float dx = dxbuf[lane];

        // edge weight for k = lane
        float z   = dx + bu0;
        float sig = 1.0f / (1.0f + __expf(-z));
        float m   = mask_ij[(size_t)n * KNB + lane];
        float w   = sig * m;
        long long gi = edge_idx[(size_t)n * KNB + lane];
        float xn  = X[gi];                 // 200KB table, L2 resident

        // node refinement partial: lane covers 4 of 128 dims
        const float4 nh  = *(const float4*)(node_h + (size_t)n * DNODE + lane * 4);
        const float4 wr4 = *(const float4*)(lds_wr + lane * 4);
        float nd = nh.x * wr4.x + nh.y * wr4.y + nh.z * wr4.z + nh.w * wr4.w;

        // single fused wave reduction: sum-w, sum-w*x, node dot
        float s0 = w, s1 = w * xn, s2 = nd;
        #pragma unroll
        for (int off = 16; off > 0; off >>= 1) {
            s0 += __shfl_xor(s0, off, 32);
            s1 += __shfl_xor(s1, off, 32);
            s2 += __shfl_xor(s2, off, 32);
        }
        if (lane == 0)
            out[n] = s1 / (s0 + 1e-6f) + s2 + br0;
    }
}

extern "C" void kernel_launch(void* const* d_in, const int* in_sizes, int n_in,
                              void* d_out, int out_size, void* d_ws, size_t ws_size,
                              hipStream_t stream) {
    (void)n_in; (void)d_ws; (void)ws_size; (void)out_size;
    const float*     X        = (const float*)d_in[0];
    const float*     node_h   = (const float*)d_in[1];
    const float*     edge_h   = (const float*)d_in[2];
    const long long* edge_idx = (const long long*)d_in[3];
    /* d_in[4] = mask_i : unused by the reference computation */
    const float*     mask_ij  = (const float*)d_in[5];
    const float*     Wu       = (const float*)d_in[6];
    const float*     bu       = (const float*)d_in[7];
    const float*     Wr       = (const float*)d_in[8];
    const float*     br       = (const float*)d_in[9];
    float* out = (float*)d_out;

    const int N = in_sizes[0];          // B*N*1 = 50000 nodes
    dim3 grid(1024), block(256);        // 8192 waves, ~6 nodes each
    graph_backbone_kernel<<<grid, block, 0, stream>>>(
        X, node_h, edge_h, edge_idx, mask_ij, Wu, bu, Wr, br, out, N);
}